// R5_90168543412477
// MI455X (gfx1250) — compile-verified
//
#include <hip/hip_runtime.h>
#include <math.h>

typedef __attribute__((ext_vector_type(2))) float v2f;
typedef __attribute__((ext_vector_type(4))) float v4f;
typedef __attribute__((ext_vector_type(8))) float v8f;

#define KC   10      // clusters
#define DF   64      // feature dim
#define EPSF 1e-8f
#define WGT  5.0f

// ---------------------------------------------------------------------------
// Kernel 1: stream feat once, WMMA f32 16x16x4 score tiles, per-block partial
// sums of (neg_sum, pos_sum, count) per cluster -> ws[block*48 + {0,16,32}+k]
// ---------------------------------------------------------------------------
__global__ __launch_bounds__(256) void cl_partial(const float* __restrict__ feat,
                                                  const int*   __restrict__ pred,
                                                  const float* __restrict__ centers,
                                                  float*       __restrict__ ws,
                                                  int n_tiles)
{
    __shared__ float lds[8 * 16 * 3];

    const int lane = threadIdx.x & 31;
    const int wave = __builtin_amdgcn_readfirstlane(threadIdx.x >> 5);
    const int half = lane >> 4;   // 0: K 0..31, 1: K 32..63
    const int r    = lane & 15;   // A: row-in-tile, B/C/D: column (cluster)

    // ---- Preload B fragments (centers^T), zero-pad columns >= KC ----------
    // Step s consumes K-slice {2s, 2s+1} (low half) / {32+2s, 32+2s+1} (high).
    v2f bfrag[16];
    if (r < KC) {
        const float* brow = centers + r * DF + half * 32;
        #pragma unroll
        for (int s = 0; s < 8; ++s) {
            v4f q = *(const v4f*)(brow + 4 * s);
            v2f b0 = {q.x, q.y};
            v2f b1 = {q.z, q.w};
            bfrag[2 * s + 0] = b0;
            bfrag[2 * s + 1] = b1;
        }
    } else {
        #pragma unroll
        for (int s = 0; s < 16; ++s) { v2f z = {0.0f, 0.0f}; bfrag[s] = z; }
    }

    float negs = 0.0f, poss = 0.0f, cnts = 0.0f;

    for (int tile = blockIdx.x * 8 + wave; tile < n_tiles; tile += gridDim.x * 8) {
        // Each lane streams a contiguous 128B run of its row (single pass -> NT).
        const float* arow = feat + (size_t)(tile * 16 + r) * DF + half * 32;
        v8f acc = {};
        #pragma unroll
        for (int s = 0; s < 8; ++s) {
            v4f q = __builtin_nontemporal_load((const v4f*)(arow + 4 * s));
            v2f a0 = {q.x, q.y};
            v2f a1 = {q.z, q.w};
            acc = __builtin_amdgcn_wmma_f32_16x16x4_f32(
                false, a0, false, bfrag[2 * s + 0], (short)0, acc, false, false);
            acc = __builtin_amdgcn_wmma_f32_16x16x4_f32(
                false, a1, false, bfrag[2 * s + 1], (short)0, acc, false, false);
        }

        // pred for the 16 rows of this tile: coalesced load + wave broadcast.
        int pmine = pred[tile * 16 + r];

        // C/D layout: lane L holds column N=L%16, rows M = v + 8*(L/16).
        #pragma unroll
        for (int v = 0; v < 8; ++v) {
            int   pv = __shfl(pmine, v + 8 * half, 32);
            float sc = acc[v] * 0.5f;                 // / TAU
            sc = fminf(10.0f, fmaxf(-10.0f, sc));     // clip
            float e = expf(sc);
            negs += e;
            if (pv == r) { poss += e; cnts += 1.0f; }
        }
    }

    // Fold the two half-columns (lane L and L+16 share cluster N).
    negs += __shfl_xor(negs, 16, 32);
    poss += __shfl_xor(poss, 16, 32);
    cnts += __shfl_xor(cnts, 16, 32);

    if (half == 0) {
        lds[(wave * 16 + r) * 3 + 0] = negs;
        lds[(wave * 16 + r) * 3 + 1] = poss;
        lds[(wave * 16 + r) * 3 + 2] = cnts;
    }
    __syncthreads();

    if (threadIdx.x < 16) {
        float n0 = 0.0f, p0 = 0.0f, c0 = 0.0f;
        #pragma unroll
        for (int w = 0; w < 8; ++w) {
            n0 += lds[(w * 16 + threadIdx.x) * 3 + 0];
            p0 += lds[(w * 16 + threadIdx.x) * 3 + 1];
            c0 += lds[(w * 16 + threadIdx.x) * 3 + 2];
        }
        float* o = ws + (size_t)blockIdx.x * 48;
        o[threadIdx.x]      = n0;
        o[16 + threadIdx.x] = p0;
        o[32 + threadIdx.x] = c0;
    }
}

// ---------------------------------------------------------------------------
// Kernel 2: reduce block partials, handle n%16 tail rows, final loss formula.
// ---------------------------------------------------------------------------
__global__ __launch_bounds__(64) void cl_final(const float* __restrict__ ws,
                                               const float* __restrict__ feat,
                                               const int*   __restrict__ pred,
                                               const float* __restrict__ centers,
                                               float*       __restrict__ out,
                                               int nblk, int n_total, int n_full)
{
    __shared__ float sneg[16], spos[16], scnt[16];
    const int t = threadIdx.x;

    if (t < 16) {
        float n = 0.0f, p = 0.0f, c = 0.0f;
        for (int b = 0; b < nblk; ++b) {
            const float* o = ws + (size_t)b * 48;
            n += o[t]; p += o[16 + t]; c += o[32 + t];
        }
        sneg[t] = n; spos[t] = p; scnt[t] = c;
    }
    __syncthreads();

    // Tail rows (n not multiple of 16): scalar path, trivial amount of work.
    if (t < 16) {
        int row = n_full + t;
        if (row < n_total) {
            int pk = pred[row];
            for (int k = 0; k < KC; ++k) {
                float d = 0.0f;
                for (int j = 0; j < DF; ++j)
                    d += feat[(size_t)row * DF + j] * centers[k * DF + j];
                float sc = fminf(10.0f, fmaxf(-10.0f, d * 0.5f));
                float e  = expf(sc);
                atomicAdd(&sneg[k], e);
                if (pk == k) { atomicAdd(&spos[k], e); atomicAdd(&scnt[k], 1.0f); }
            }
        }
    }
    __syncthreads();

    if (t == 0) {
        float total = 0.0f; int nvalid = 0;
        for (int k = 0; k < KC; ++k) {
            float cnt = scnt[k];
            float pos = spos[k] / fmaxf(cnt, 1.0f);
            float neg = sneg[k] / (float)n_total;
            float lt  = -logf(pos / (neg + EPSF));
            bool valid = (cnt > 0.0f) && (cnt < (float)n_total) &&
                         (pos > EPSF) && (neg > EPSF) && __builtin_isfinite(lt);
            if (valid) { total += lt; ++nvalid; }
        }
        out[0] = (nvalid > 0 ? total / (float)nvalid : 0.0f) * WGT;
    }
}

// ---------------------------------------------------------------------------
extern "C" void kernel_launch(void* const* d_in, const int* in_sizes, int n_in,
                              void* d_out, int out_size, void* d_ws, size_t ws_size,
                              hipStream_t stream)
{
    const float* feat    = (const float*)d_in[0];
    const int*   pred    = (const int*)d_in[1];
    const float* centers = (const float*)d_in[2];
    float*       ws      = (float*)d_ws;
    float*       out     = (float*)d_out;

    const int n       = in_sizes[1];     // N_CELLS (pred length)
    const int n_tiles = n / 16;
    const int n_full  = n_tiles * 16;

    int nblk = 512;
    if ((size_t)nblk * 48 * sizeof(float) > ws_size) {
        nblk = (int)(ws_size / (48 * sizeof(float)));
        if (nblk < 1) nblk = 1;
    }

    cl_partial<<<dim3(nblk), dim3(256), 0, stream>>>(feat, pred, centers, ws, n_tiles);
    cl_final<<<dim3(1), dim3(64), 0, stream>>>(ws, feat, pred, centers, out,
                                               nblk, n, n_full);
}